// QuanvolutionHybridClassifier_65481071400812
// MI455X (gfx1250) — compile-verified
//
#include <hip/hip_runtime.h>
#include <math.h>

typedef __attribute__((ext_vector_type(2))) float v2f;
typedef __attribute__((ext_vector_type(8))) float v8f;

// ---------------------------------------------------------------------------
// Stage 1: build the 16x16 complex unitary U = G7 * ... * G0 from theta[8].
// One block of 256 threads; thread t = row*16 + col owns element U[row][col].
// Gate list (wire0 <-> amp bit 3):
//   rx(w0) ry(w1) rz(w3) crx(w0,w2) ry(w3) rx(w2) crx(w1,w3) rz(w0)
// ---------------------------------------------------------------------------
__global__ void build_unitary_kernel(const float* __restrict__ theta,
                                     float* __restrict__ uR,
                                     float* __restrict__ uI) {
    __shared__ float cr[256], ci[256];
    const int t   = threadIdx.x;
    const int row = t >> 4;
    const int col = t & 15;

    cr[t] = (row == col) ? 1.0f : 0.0f;
    ci[t] = 0.0f;
    __syncthreads();

#pragma unroll
    for (int k = 0; k < 8; ++k) {
        // compile-time gate metadata (loop is fully unrolled)
        constexpr int G_TYPE[8]  = {0, 1, 2, 3, 1, 0, 3, 2};  // 0=rx 1=ry 2=rz 3=crx
        constexpr int G_TMASK[8] = {8, 4, 1, 2, 1, 2, 1, 8};  // target bit mask
        constexpr int G_CMASK[8] = {0, 0, 0, 8, 0, 0, 4, 0};  // control bit mask
        const int ty = G_TYPE[k], tm = G_TMASK[k], cm = G_CMASK[k];

        float th = theta[k];
        float s, c;
        __sincosf(0.5f * th, &s, &c);
        const int t2 = t ^ (tm << 4);   // partner row, same column

        float xr = cr[t],  xi = ci[t];
        float yr = cr[t2], yi = ci[t2];
        float oR = xr, oI = xi;

        if (ty == 0) {                       // rx: [[c,-is],[-is,c]]
            oR = c * xr + s * yi;
            oI = c * xi - s * yr;
        } else if (ty == 1) {                // ry: [[c,-s],[s,c]]
            float sg = (row & tm) ? s : -s;
            oR = c * xr + sg * yr;
            oI = c * xi + sg * yi;
        } else if (ty == 2) {                // rz: diag(e^{-it/2}, e^{+it/2})
            float sg = (row & tm) ? 1.0f : -1.0f;
            oR = xr * c - sg * xi * s;
            oI = xi * c + sg * xr * s;
        } else {                             // crx: rx on target if ctrl bit set
            if (row & cm) {
                oR = c * xr + s * yi;
                oI = c * xi - s * yr;
            }
        }
        __syncthreads();
        cr[t] = oR;
        ci[t] = oI;
        __syncthreads();
    }
    uR[t] = cr[t];   // U[row][col] : row = output amp, col = input amp
    uI[t] = ci[t];
}

// ---------------------------------------------------------------------------
// Stage 2: per-wave tile of 16 patches.
//   D(16 amps_out x 16 patches) = U(16x16) x Psi(16x16): 4 K-chunks of
//   V_WMMA_F32_16X16X4_F32 for each of Ur and Ui (8 WMMAs per tile).
//   B-fragment element k = 4*cch + 2*hi + e is built directly from the
//   per-wire cos/sin factors (no psi[] register array -> no select chains):
//     psi[k] = q0[cch>>1] * q1[cch&1] * q2[hi] * q3[e]
//   probs = dR^2 + dI^2 stays in D layout; PauliZ reduction = compile-time
//   +-signs over the 8 accumulator VGPRs + one shfl_xor(16) for amp bit 3.
// ---------------------------------------------------------------------------
__global__ void quanv_kernel(const float* __restrict__ x,
                             const float* __restrict__ uR,
                             const float* __restrict__ uI,
                             float* __restrict__ feats,
                             int npatch) {
    __shared__ float sUR[256], sUI[256];
    const int tid = threadIdx.x;
    sUR[tid] = uR[tid];
    sUI[tid] = uI[tid];
    __syncthreads();

    const int wave = tid >> 5;
    const int lane = tid & 31;
    const int hi   = lane >> 4;        // lane half
    const int p    = lane & 15;        // patch within tile (= N column)
    const int tile = blockIdx.x * 8 + wave;
    if (tile * 16 >= npatch) return;   // wave-uniform: EXEC stays all-ones

    int g  = tile * 16 + p;
    int gc = (g < npatch) ? g : (npatch - 1);
    const int b  = gc / 196;
    const int pi = gc % 196;
    const int pr = pi / 14, pc = pi % 14;

    const float* img = x + b * 784;
    const float a0 = img[(2 * pr)     * 28 + 2 * pc];
    const float a1 = img[(2 * pr)     * 28 + 2 * pc + 1];
    const float a2 = img[(2 * pr + 1) * 28 + 2 * pc];
    const float a3 = img[(2 * pr + 1) * 28 + 2 * pc + 1];

    float c0, s0, c1, s1, c2, s2, c3, s3;
    __sincosf(0.5f * a0, &s0, &c0);
    __sincosf(0.5f * a1, &s1, &c1);
    __sincosf(0.5f * a2, &s2, &c2);
    __sincosf(0.5f * a3, &s3, &c3);

    // lane-half factor (amp bit1 = hi) folded once; bit0 = element (.x/.y)
    const float w2 = hi ? s2 : c2;     // single cndmask
    const float px = w2 * c3;          // e = 0
    const float py = w2 * s3;          // e = 1
    // chunk prefixes (amp bits 3,2 = cch bits 1,0) -- all compile-time picks
    const float pf0 = c0 * c1;
    const float pf1 = c0 * s1;
    const float pf2 = s0 * c1;
    const float pf3 = s0 * s1;

    v8f dR = {0.f, 0.f, 0.f, 0.f, 0.f, 0.f, 0.f, 0.f};
    v8f dI = {0.f, 0.f, 0.f, 0.f, 0.f, 0.f, 0.f, 0.f};
    const int i    = p;                // A row (output amp) = lane % 16
    const int koff = 2 * hi;           // lane-half K offset (address math only)

    auto do_chunk = [&](int base, float pf) {
        v2f aR, aI, bp;
        const float2 tR = *(const float2*)&sUR[i * 16 + base + koff];
        const float2 tI = *(const float2*)&sUI[i * 16 + base + koff];
        aR.x = tR.x;  aR.y = tR.y;
        aI.x = tI.x;  aI.y = tI.y;
        bp.x = pf * px;
        bp.y = pf * py;
        dR = __builtin_amdgcn_wmma_f32_16x16x4_f32(false, aR, false, bp,
                                                   (short)0, dR, false, false);
        dI = __builtin_amdgcn_wmma_f32_16x16x4_f32(false, aI, false, bp,
                                                   (short)0, dI, false, false);
    };
    do_chunk(0,  pf0);
    do_chunk(4,  pf1);
    do_chunk(8,  pf2);
    do_chunk(12, pf3);

    // probs + signed reductions (Walsh signs per amp bit)
    float m0 = 0.f, m1 = 0.f, m2 = 0.f, m3 = 0.f;
#pragma unroll
    for (int r = 0; r < 8; ++r) {
        const float v = dR[r] * dR[r] + dI[r] * dI[r];
        m0 += v;                               // amp bit3 handled via half
        m1 += ((r >> 2) & 1) ? -v : v;         // amp bit2
        m2 += ((r >> 1) & 1) ? -v : v;         // amp bit1
        m3 += (r & 1) ? -v : v;                // amp bit0
    }
    if (hi) m0 = -m0;
    m0 += __shfl_xor(m0, 16, 32);
    m1 += __shfl_xor(m1, 16, 32);
    m2 += __shfl_xor(m2, 16, 32);
    m3 += __shfl_xor(m3, 16, 32);

    if (hi == 0 && g < npatch) {
        float* f = feats + g * 4;
        f[0] = m0; f[1] = m1; f[2] = m2; f[3] = m3;
    }
}

// ---------------------------------------------------------------------------
// Stage 3: logits = feats(B x 784) @ W^T(784 x 10) + b, then log_softmax.
// One wave per 16 batch rows; N = classes (10 used, 6 zero-padded);
// K = 784 -> 196 chained f32 WMMAs.  Softmax via 16-lane shfl reductions.
// ---------------------------------------------------------------------------
__global__ void head_kernel(const float* __restrict__ feats,
                            const float* __restrict__ W,
                            const float* __restrict__ bias,
                            float* __restrict__ out,
                            int B) {
    const int tid  = threadIdx.x;
    const int wave = tid >> 5;
    const int lane = tid & 31;
    const int hi   = lane >> 4;
    const int n    = lane & 15;        // class column (valid < 10)
    const int tile = blockIdx.x * 8 + wave;
    if (tile * 16 >= B) return;        // wave-uniform

    const int rowbase = tile * 16;
    int arow = rowbase + n;            // A row for this lane
    if (arow >= B) arow = B - 1;
    const bool wvalid = (n < 10);
    // fold the lane-half K offset into the base pointers; both rows are
    // 8-byte aligned at even k0, so use b64 (float2) loads
    const float* frow = feats + (size_t)arow * 784 + 2 * hi;
    const float* wrow = W + (wvalid ? n : 0) * 784 + 2 * hi;

    v8f d = {0.f, 0.f, 0.f, 0.f, 0.f, 0.f, 0.f, 0.f};
    for (int cch = 0; cch < 196; ++cch) {
        const float2 fa = *(const float2*)(frow + cch * 4);
        const float2 wb = *(const float2*)(wrow + cch * 4);
        v2f a, bf;
        a.x  = fa.x;
        a.y  = fa.y;
        bf.x = wvalid ? wb.x : 0.0f;
        bf.y = wvalid ? wb.y : 0.0f;
        d = __builtin_amdgcn_wmma_f32_16x16x4_f32(false, a, false, bf,
                                                  (short)0, d, false, false);
    }

    const float bn = wvalid ? bias[n] : 0.0f;
#pragma unroll
    for (int r = 0; r < 8; ++r) {
        const float v  = d[r] + bn;
        float mv = wvalid ? v : -3.0e38f;
        mv = fmaxf(mv, __shfl_xor(mv, 1, 32));
        mv = fmaxf(mv, __shfl_xor(mv, 2, 32));
        mv = fmaxf(mv, __shfl_xor(mv, 4, 32));
        mv = fmaxf(mv, __shfl_xor(mv, 8, 32));
        float e = wvalid ? __expf(v - mv) : 0.0f;
        e += __shfl_xor(e, 1, 32);
        e += __shfl_xor(e, 2, 32);
        e += __shfl_xor(e, 4, 32);
        e += __shfl_xor(e, 8, 32);
        const float ls  = v - mv - __logf(e);
        const int   row = rowbase + r + 8 * hi;
        if (wvalid && row < B) out[row * 10 + n] = ls;
    }
}

// ---------------------------------------------------------------------------
extern "C" void kernel_launch(void* const* d_in, const int* in_sizes, int n_in,
                              void* d_out, int out_size, void* d_ws, size_t ws_size,
                              hipStream_t stream) {
    const float* x     = (const float*)d_in[0];
    const float* theta = (const float*)d_in[1];
    const float* W     = (const float*)d_in[2];
    const float* bias  = (const float*)d_in[3];
    float*       out   = (float*)d_out;

    const int B      = in_sizes[0] / 784;   // 2048
    const int npatch = B * 196;             // 401408

    float* uR    = (float*)d_ws;            // 256 floats
    float* uI    = uR + 256;                // 256 floats
    float* feats = uI + 256;                // npatch*4 floats (~6.4 MB)

    build_unitary_kernel<<<1, 256, 0, stream>>>(theta, uR, uI);

    const int tiles  = (npatch + 15) / 16;  // 25088
    const int blocks = (tiles + 7) / 8;     // 3136
    quanv_kernel<<<blocks, 256, 0, stream>>>(x, uR, uI, feats, npatch);

    const int rtiles = (B + 15) / 16;       // 128
    head_kernel<<<(rtiles + 7) / 8, 256, 0, stream>>>(feats, W, bias, out, B);
}